// TSFeatureExtractor_41772851921486
// MI455X (gfx1250) — compile-verified
//
#include <hip/hip_runtime.h>
#include <hip/hip_bf16.h>
#include <math.h>

// ---------------------------------------------------------------------------
// TS feature extractor for MI455X (gfx1250, wave32).
//   x: [B=128, T=4096, V=32] f32  ->  out: [B, 32] f32
// Kernel 1: streaming moments/ACF (memory-bound; 2 passes, both L2-resident).
// Kernel 2: rFFT-4096 via 64x64 Cooley-Tukey = 6 real 64x64x64 matmuls per
//           series, executed with V_WMMA_F32_16X16X4_F32 (f32 precision).
//           A-fragments are hoisted out of the tile loop into registers to
//           cut LDS traffic / address VALU feeding the matrix pipe.
// Kernel 3: mean/std over V + clip.
// ---------------------------------------------------------------------------

#define TS_B 128
#define TS_T 4096
#define TS_V 32
#define TS_NF 16
#define TS_PI 3.14159265358979323846f

typedef float v2f __attribute__((ext_vector_type(2)));
typedef float v8f __attribute__((ext_vector_type(8)));

__device__ __forceinline__ v8f wmma4(v2f a, v2f b, v8f c) {
  // D = A(16x4 f32) * B(4x16 f32) + C(16x16 f32)
  return __builtin_amdgcn_wmma_f32_16x16x4_f32(
      /*neg_a=*/false, a, /*neg_b=*/false, b,
      /*c_mod=*/(short)0, c, /*reuse_a=*/false, /*reuse_b=*/false);
}

// ---------------------------------------------------------------------------
// Kernel 1: per-(b,v) moments, extrema, slope, ACF lags 1..6.
// Block = 256 threads: thread (v = tid&31, g = tid>>5) handles a contiguous
// 512-sample chunk of series (b, v). Lanes sweep v -> fully coalesced loads.
// ---------------------------------------------------------------------------
__global__ __launch_bounds__(256) void ts_moments_kernel(
    const float* __restrict__ x, float* __restrict__ feats) {
  const int b = blockIdx.x;
  const int tid = (int)threadIdx.x;
  const int v = tid & (TS_V - 1);
  const int g = tid >> 5;          // 0..7
  const int CH = TS_T / 8;         // 512
  const int t0 = g * CH;
  const float* __restrict__ xb = x + (size_t)b * TS_T * TS_V;

  __shared__ float redA[4][8][TS_V];
  __shared__ float redB[9][8][TS_V];
  __shared__ float s_mean[TS_V];

  // ---- pass 1: sum, sum(x*t), min, max
  float s = 0.f, st = 0.f, mn = 3.4e38f, mx = -3.4e38f;
  for (int t = t0; t < t0 + CH; ++t) {
    float xt = xb[(size_t)t * TS_V + v];
    s += xt;
    st += xt * (float)t;
    mn = fminf(mn, xt);
    mx = fmaxf(mx, xt);
  }
  redA[0][g][v] = s; redA[1][g][v] = st; redA[2][g][v] = mn; redA[3][g][v] = mx;
  __syncthreads();
  if (tid < TS_V) {
    float S = 0.f, ST = 0.f, MN = 3.4e38f, MX = -3.4e38f;
    for (int k = 0; k < 8; ++k) {
      S += redA[0][k][tid];
      ST += redA[1][k][tid];
      MN = fminf(MN, redA[2][k][tid]);
      MX = fmaxf(MX, redA[3][k][tid]);
    }
    float mean = S / (float)TS_T;
    s_mean[tid] = mean;
    const double tmean = ((double)TS_T - 1.0) * 0.5;
    const double den = (double)TS_T * ((double)TS_T * (double)TS_T - 1.0) / 12.0;
    float slope = (float)(((double)ST - tmean * (double)S) / den);
    float* f = feats + ((size_t)b * TS_V + tid) * TS_NF;
    f[0] = mean; f[4] = MN; f[5] = MX; f[6] = slope;
  }
  __syncthreads();

  // ---- pass 2: centered powers + ACF cross-terms (ring buffer of xc[t-1..t-6])
  const float m = s_mean[v];
  float rb[6];
#pragma unroll
  for (int L = 1; L <= 6; ++L) {
    int tp = t0 - L;
    rb[L - 1] = (tp >= 0) ? (xb[(size_t)tp * TS_V + v] - m) : 0.f;
  }
  float m2 = 0.f, m3 = 0.f, m4 = 0.f;
  float ac[6] = {0.f, 0.f, 0.f, 0.f, 0.f, 0.f};
  for (int t = t0; t < t0 + CH; ++t) {
    float xc = xb[(size_t)t * TS_V + v] - m;
    float x2 = xc * xc;
    m2 += x2; m3 += x2 * xc; m4 += x2 * x2;
#pragma unroll
    for (int L = 0; L < 6; ++L) ac[L] += xc * rb[L];
    rb[5] = rb[4]; rb[4] = rb[3]; rb[3] = rb[2];
    rb[2] = rb[1]; rb[1] = rb[0]; rb[0] = xc;
  }
  redB[0][g][v] = m2; redB[1][g][v] = m3; redB[2][g][v] = m4;
#pragma unroll
  for (int L = 0; L < 6; ++L) redB[3 + L][g][v] = ac[L];
  __syncthreads();
  if (tid < TS_V) {
    float q[9];
    for (int j = 0; j < 9; ++j) {
      float a = 0.f;
      for (int k = 0; k < 8; ++k) a += redB[j][k][tid];
      q[j] = a;
    }
    float var = q[0] / (float)TS_T;
    float sd = sqrtf(var + 1e-8f);
    float inv3 = 1.f / (sd * sd * sd);
    float* f = feats + ((size_t)b * TS_V + tid) * TS_NF;
    f[1] = sd;
    f[2] = (q[1] / (float)TS_T) * inv3;        // skew = E[z^3]
    f[3] = (q[2] / (float)TS_T) * inv3 / sd;   // kurt = E[z^4]
    float invd = 1.f / (q[0] + 1e-8f);
#pragma unroll
    for (int L = 0; L < 6; ++L) f[10 + L] = q[3 + L] * invd;
  }
}

// ---------------------------------------------------------------------------
// Kernel 2: DFT-4096 per series via 64x64 split, f32 WMMA, then top-3 |X_k|/T
// over k = 1..2048. One workgroup (128 thr = 4 waves) per (b,v) series.
//   t = t1*64 + t2, k = k1 + 64*k2
//   inner[k1,t2] = sum_t1 W64[k1*t1] * x[t1,t2]        (2 real matmuls, WMMA)
//   tw[k1,t2]    = inner * e^{-2pi i k1 t2 / 4096}      (elementwise)
//   X[k1,k2]     = sum_t2 tw[k1,t2] * W64[k2*t2]        (4 real matmuls, WMMA)
// Wave w owns k1-block [16w, 16w+16) for both stages. A-fragments depend only
// on (k1-block, kstep), so they are loaded ONCE into registers per stage and
// reused across all 4 output-tile columns.
// ---------------------------------------------------------------------------
__global__ __launch_bounds__(128) void ts_fft_topk_kernel(
    const float* __restrict__ x, float* __restrict__ feats) {
  const int bv = (int)blockIdx.x;
  const int b = bv >> 5;
  const int v = bv & 31;
  const int tid = (int)threadIdx.x;
  const int lane = tid & 31;
  const int w = tid >> 5;  // wave 0..3

  __shared__ float sX[64 * 64];     // [t1][t2]
  __shared__ float sTwRe[64 * 64];  // [k1][t2]
  __shared__ float sTwIm[64 * 64];
  __shared__ float sWre[64];        // e^{-2pi i j/64}
  __shared__ float sWim[64];
  __shared__ float sTop[128 * 3];

  if (tid < 64) {
    float ang = (-2.f * TS_PI / 64.f) * (float)tid;
    float sv, cv;
    __sincosf(ang, &sv, &cv);
    sWre[tid] = cv;
    sWim[tid] = sv;
  }
  const float* __restrict__ xs = x + (size_t)b * TS_T * TS_V + v;
  for (int i = tid; i < 4096; i += 128) sX[i] = xs[(size_t)i * TS_V];
  __syncthreads();

  const int lm = lane & 15;  // per ISA A/B/C lane layouts
  const int lh = lane >> 4;
  const int mrow = w * 16 + lm;  // k1 row owned by this wave (A "m" index)

  // ---- Stage 1: inner = W64 x X, then twiddle -> sTw*
  {
    // Hoist A-fragments (DFT64 cos/sin rows) into registers: reused for all tn.
    v2f aReF[16], aImF[16];
#pragma unroll
    for (int ks = 0; ks < 16; ++ks) {
      const int kb = ks * 4 + 2 * lh;  // t1 for vgpr j=0
      int p0 = (mrow * kb) & 63;
      int p1 = (mrow * (kb + 1)) & 63;
      aReF[ks].x = sWre[p0]; aReF[ks].y = sWre[p1];
      aImF[ks].x = sWim[p0]; aImF[ks].y = sWim[p1];
    }
    for (int tn = 0; tn < 4; ++tn) {
      const int ncol = tn * 16 + lm;  // t2 (B col)
      v8f accRe = {0.f, 0.f, 0.f, 0.f, 0.f, 0.f, 0.f, 0.f};
      v8f accIm = {0.f, 0.f, 0.f, 0.f, 0.f, 0.f, 0.f, 0.f};
#pragma unroll
      for (int ks = 0; ks < 16; ++ks) {
        const int kb = ks * 4 + 2 * lh;
        v2f bX;
        bX.x = sX[kb * 64 + ncol];
        bX.y = sX[(kb + 1) * 64 + ncol];
        accRe = wmma4(aReF[ks], bX, accRe);
        accIm = wmma4(aImF[ks], bX, accIm);
      }
#pragma unroll
      for (int r = 0; r < 8; ++r) {
        int k1 = w * 16 + r + 8 * lh;  // C/D layout: M = r + 8*(lane>>4)
        int t2 = ncol;
        float ang = (-2.f * TS_PI / 4096.f) * (float)(k1 * t2);
        float sv, cv;
        __sincosf(ang, &sv, &cv);
        float re = accRe[r], im = accIm[r];
        sTwRe[k1 * 64 + t2] = re * cv - im * sv;
        sTwIm[k1 * 64 + t2] = re * sv + im * cv;
      }
    }
  }
  __syncthreads();

  // ---- Stage 2: complex matmul with W64, magnitudes, per-lane top-3
  float top0 = -1.f, top1 = -1.f, top2 = -1.f;
  {
    // Hoist A-fragments (twiddled intermediate rows) into registers.
    v2f aRF[16], aIF[16];
#pragma unroll
    for (int ks = 0; ks < 16; ++ks) {
      const int kb = ks * 4 + 2 * lh;  // t2
      aRF[ks].x = sTwRe[mrow * 64 + kb];
      aRF[ks].y = sTwRe[mrow * 64 + kb + 1];
      aIF[ks].x = sTwIm[mrow * 64 + kb];
      aIF[ks].y = sTwIm[mrow * 64 + kb + 1];
    }
    for (int tn = 0; tn < 4; ++tn) {
      const int ncol = tn * 16 + lm;  // k2
      v8f accRe = {0.f, 0.f, 0.f, 0.f, 0.f, 0.f, 0.f, 0.f};
      v8f accIm = {0.f, 0.f, 0.f, 0.f, 0.f, 0.f, 0.f, 0.f};
#pragma unroll
      for (int ks = 0; ks < 16; ++ks) {
        const int kb = ks * 4 + 2 * lh;
        int q0 = (ncol * kb) & 63;
        int q1 = (ncol * (kb + 1)) & 63;
        v2f bC; bC.x = sWre[q0]; bC.y = sWre[q1];
        v2f bS; bS.x = sWim[q0]; bS.y = sWim[q1];
        v2f aIn; aIn.x = -aIF[ks].x; aIn.y = -aIF[ks].y;
        // (aR + i aI)(bC + i bS) = (aR*bC - aI*bS) + i(aR*bS + aI*bC)
        accRe = wmma4(aRF[ks], bC, accRe);
        accRe = wmma4(aIn, bS, accRe);
        accIm = wmma4(aRF[ks], bS, accIm);
        accIm = wmma4(aIF[ks], bC, accIm);
      }
#pragma unroll
      for (int r = 0; r < 8; ++r) {
        int k1 = w * 16 + r + 8 * lh;
        int k2 = ncol;
        int k = k1 + 64 * k2;
        if (k >= 1 && k <= 2048) {  // rfft bins 1..2048 (|X_k| symmetric)
          float re = accRe[r], im = accIm[r];
          float mag = sqrtf(re * re + im * im) * (1.f / (float)TS_T);
          if (mag > top0)      { top2 = top1; top1 = top0; top0 = mag; }
          else if (mag > top1) { top2 = top1; top1 = mag; }
          else if (mag > top2) { top2 = mag; }
        }
      }
    }
  }
  sTop[tid * 3 + 0] = top0;
  sTop[tid * 3 + 1] = top1;
  sTop[tid * 3 + 2] = top2;
  __syncthreads();
  if (tid == 0) {
    float a0 = -1.f, a1 = -1.f, a2 = -1.f;
    for (int i = 0; i < 128 * 3; ++i) {
      float g = sTop[i];
      if (g > a0)      { a2 = a1; a1 = a0; a0 = g; }
      else if (g > a1) { a2 = a1; a1 = g; }
      else if (g > a2) { a2 = g; }
    }
    float* f = feats + ((size_t)b * TS_V + v) * TS_NF;
    f[7] = a0; f[8] = a1; f[9] = a2;
  }
}

// ---------------------------------------------------------------------------
// Kernel 3: mean/std over V per feature slot, clip to +-5.
// ---------------------------------------------------------------------------
__global__ __launch_bounds__(32) void ts_finalize_kernel(
    const float* __restrict__ feats, float* __restrict__ out) {
  const int b = blockIdx.x;
  const int j = (int)threadIdx.x;
  if (j < TS_NF) {
    float s = 0.f, ss = 0.f;
    for (int vv = 0; vv < TS_V; ++vv) {
      float f = feats[((size_t)b * TS_V + vv) * TS_NF + j];
      s += f;
      ss += f * f;
    }
    float mean = s / (float)TS_V;
    float var = ss / (float)TS_V - mean * mean;
    float sd = sqrtf(fmaxf(var, 0.f));
    out[(size_t)b * 32 + j] = fminf(fmaxf(mean, -5.f), 5.f);
    out[(size_t)b * 32 + 16 + j] = fminf(fmaxf(sd, -5.f), 5.f);
  }
}

extern "C" void kernel_launch(void* const* d_in, const int* in_sizes, int n_in,
                              void* d_out, int out_size, void* d_ws, size_t ws_size,
                              hipStream_t stream) {
  (void)in_sizes; (void)n_in; (void)out_size; (void)ws_size;
  const float* x = (const float*)d_in[0];
  float* out = (float*)d_out;
  float* feats = (float*)d_ws;  // [B, V, 16] f32 = 256 KB scratch

  hipLaunchKernelGGL(ts_moments_kernel, dim3(TS_B), dim3(256), 0, stream, x, feats);
  hipLaunchKernelGGL(ts_fft_topk_kernel, dim3(TS_B * TS_V), dim3(128), 0, stream, x, feats);
  hipLaunchKernelGGL(ts_finalize_kernel, dim3(TS_B), dim3(32), 0, stream, feats, out);
}